// TextualGraph_15144054686140
// MI455X (gfx1250) — compile-verified
//
#include <hip/hip_runtime.h>
#include <math.h>

// ---- problem constants (match reference) ----
#define N_IMAGE   128
#define N_CAPTION 32
#define N_REGION  36
#define N_WORD    24
#define EMBED     1024
#define NUM_BLOCK 32
#define BLOCKW    32      // EMBED / NUM_BLOCK
#define N_KERNELS 8
#define HID       64
#define LAMBDA_F  4.0f
#define EPSF      1e-8f

typedef float v2f __attribute__((ext_vector_type(2)));
typedef float v8f __attribute__((ext_vector_type(8)));

// ---- workspace layout (float offsets) ----
#define OFF_ATTN  ((size_t)0)                            // 32*128*24*36 = 3538944
#define OFF_TNODE (OFF_ATTN  + (size_t)N_CAPTION*N_IMAGE*N_WORD*N_REGION)   // +3145728
#define OFF_KWA   (OFF_TNODE + (size_t)N_CAPTION*N_IMAGE*N_WORD*NUM_BLOCK)  // +147456
#define OFF_QN    (OFF_KWA   + (size_t)N_CAPTION*N_WORD*N_WORD*N_KERNELS)   // +24576
#define OFF_W1N   (OFF_QN    + (size_t)N_CAPTION*N_WORD*NUM_BLOCK)          // +4096
#define OFF_W2N   (OFF_W1N   + (size_t)HID*HID)                             // +64

__device__ __forceinline__ float half16_sum(float v) {
  // reduction within a 16-lane half of a wave32 (xor masks stay in-half)
  v += __shfl_xor(v, 1, 32);
  v += __shfl_xor(v, 2, 32);
  v += __shfl_xor(v, 4, 32);
  v += __shfl_xor(v, 8, 32);
  return v;
}

// ======================================================================
// Kernel 1: weight-norm of out1 / out2 linear layers
// ======================================================================
__global__ void weight_prep_kernel(const float* __restrict__ w1v,
                                   const float* __restrict__ w1g,
                                   const float* __restrict__ w2v,
                                   const float* __restrict__ w2g,
                                   float* __restrict__ W1n,
                                   float* __restrict__ W2n) {
  __shared__ float sc2;
  const int t = threadIdx.x;  // 64 threads
  float ss = 0.f;
  for (int u = 0; u < HID; ++u) { float v = w1v[t * HID + u]; ss += v * v; }
  const float scale = w1g[t] / sqrtf(ss);
  for (int u = 0; u < HID; ++u) W1n[t * HID + u] = w1v[t * HID + u] * scale;
  if (t == 0) {
    float s2 = 0.f;
    for (int u = 0; u < HID; ++u) s2 += w2v[u] * w2v[u];
    sc2 = w2g[0] / sqrtf(s2);
  }
  __syncthreads();
  W2n[t] = w2v[t] * sc2;
}

// ======================================================================
// Kernel 2: per-caption prep: words_sim softmax, adjacency, kwa, q-norms
// ======================================================================
__global__ void caption_prep_kernel(const float* __restrict__ caps,
                                    const int*   __restrict__ depends,
                                    const float* __restrict__ mean_k,
                                    const float* __restrict__ prec_k,
                                    float* __restrict__ kwa,
                                    float* __restrict__ qn) {
  __shared__ float ws[N_WORD][N_WORD];
  __shared__ float adj[N_WORD][N_WORD];
  __shared__ float rn[N_WORD];
  const int c   = blockIdx.x;
  const int tid = threadIdx.x;
  const float* cap = caps + (size_t)c * N_WORD * EMBED;

  // raw self-attention scores * lambda
  for (int p = tid; p < N_WORD * N_WORD; p += blockDim.x) {
    const int i = p / N_WORD, j = p % N_WORD;
    const float* ri = cap + (size_t)i * EMBED;
    const float* rj = cap + (size_t)j * EMBED;
    float d = 0.f;
    for (int k = 0; k < EMBED; ++k) d += ri[k] * rj[k];
    ws[i][j]  = LAMBDA_F * d;
    adj[i][j] = 0.f;
  }
  __syncthreads();
  // row softmax
  if (tid < N_WORD) {
    float mx = -1e30f;
    for (int j = 0; j < N_WORD; ++j) mx = fmaxf(mx, ws[tid][j]);
    float s = 0.f;
    for (int j = 0; j < N_WORD; ++j) { float e = expf(ws[tid][j] - mx); ws[tid][j] = e; s += e; }
    const float inv = 1.f / s;
    for (int j = 0; j < N_WORD; ++j) ws[tid][j] *= inv;
  }
  __syncthreads();
  // sparse symmetric adjacency (dependency pair 0 skipped)
  if (tid >= 1 && tid < N_WORD - 1) {
    const int r  = depends[((size_t)c * (N_WORD - 1) + tid) * 2 + 0];
    const int cc = depends[((size_t)c * (N_WORD - 1) + tid) * 2 + 1];
    if (r >= 0 && r < N_WORD && cc >= 0 && cc < N_WORD) { adj[r][cc] = 1.f; adj[cc][r] = 1.f; }
  }
  __syncthreads();
  if (tid < N_WORD) adj[tid][tid] += 1.f;   // + eye
  __syncthreads();
  // row norms of adj*words_sim
  if (tid < N_WORD) {
    float ss = 0.f;
    for (int j = 0; j < N_WORD; ++j) { float v = adj[tid][j] * ws[tid][j]; ss += v * v; }
    rn[tid] = sqrtf(ss) + EPSF;
  }
  __syncthreads();
  // gaussian kernel weights (normalized over k), pre-multiplied by adjacency
  for (int p = tid; p < N_WORD * N_WORD; p += blockDim.x) {
    const int i = p / N_WORD, j = p % N_WORD;
    const float a = adj[i][j] * ws[i][j] / rn[i];
    float e[N_KERNELS]; float s = 0.f;
    for (int k = 0; k < N_KERNELS; ++k) {
      const float d  = a - mean_k[k];
      const float pk = prec_k[k];
      e[k] = expf(-0.5f * d * d / (1e-14f + pk * pk));
      s += e[k];
    }
    float* dst = kwa + ((size_t)(c * N_WORD + i) * N_WORD + j) * N_KERNELS;
    for (int k = 0; k < N_KERNELS; ++k) dst[k] = adj[i][j] * e[k] / s;
  }
  // per-block q norms
  for (int p = tid; p < N_WORD * NUM_BLOCK; p += blockDim.x) {
    const int w = p / NUM_BLOCK, kb = p % NUM_BLOCK;
    const float* q = cap + (size_t)w * EMBED + kb * BLOCKW;
    float ss = 0.f;
    for (int t2 = 0; t2 < BLOCKW; ++t2) ss += q[t2] * q[t2];
    qn[(size_t)(c * N_WORD + w) * NUM_BLOCK + kb] = sqrtf(ss);
  }
}

// ======================================================================
// Kernel 3: attn = images_flat @ cap^T  (fp32 WMMA 16x16x4, K=1024)
//   B (caption) staged per-block into LDS via async-to-LDS DMA, double
//   buffered so chunk i+1's DMA overlaps chunk i's WMMA loop.
//   Fused leaky-ReLU + l2norm over w; stored transposed (c,b,w,r).
// grid: 32 captions * 36 groups, block = 256 (8 waves, 1 m-tile each)
// ======================================================================
#define KC      128                         // K-chunk (floats)
#define NCHUNK  (EMBED / KC)                // 8
#define NXFER   (N_WORD * KC / 4)           // 768 b128 transfers per chunk

__global__ void attn_gemm_kernel(const float* __restrict__ images,
                                 const float* __restrict__ captions,
                                 float* __restrict__ attn) {
  __shared__ float bsm[2][N_WORD * KC];     // 2 x 12 KB caption chunk
  const int tid  = threadIdx.x;
  const int wv   = tid >> 5;
  const int lane = tid & 31;
  const int c    = blockIdx.x / (288 / 8);
  const int grp  = blockIdx.x % (288 / 8);
  const int m0   = (grp * 8 + wv) * 16;     // row tile of (b*36+r)
  const int lo = lane & 15, hi = lane >> 4;

  const float* Arow = images + (size_t)(m0 + lo) * EMBED + 2 * hi;
  const float* cap  = captions + (size_t)c * N_WORD * EMBED;
  const int n2c     = (16 + lo < N_WORD) ? (16 + lo) : (N_WORD - 1); // clamp pad cols
  const unsigned smem_base = (unsigned)(size_t)&bsm[0][0];  // flat[31:0] == LDS offset

  // issue one chunk's async global->LDS DMA (B128 = 4 floats / transfer)
  auto issue_chunk = [&](int kc, int buf) {
    for (int e = tid; e < NXFER; e += 256) {
      const int row = e >> 5;                  // 32 transfers per 128-float row
      const int q   = e & 31;
      const unsigned ldsa =
          smem_base + (unsigned)((buf * N_WORD * KC + row * KC + q * 4) * 4);
      const unsigned long long ga =
          (unsigned long long)(size_t)(cap + (size_t)row * EMBED + kc + q * 4);
      asm volatile("global_load_async_to_lds_b128 %0, %1, off"
                   :: "v"(ldsa), "v"(ga) : "memory");
    }
  };

  v8f acc0 = {}; v8f acc1 = {};
  issue_chunk(0, 0);
  for (int ch = 0; ch < NCHUNK; ++ch) {
    asm volatile("s_wait_asynccnt 0x0" ::: "memory");  // my chunk-ch DMAs done
    __syncthreads();                                   // everyone's DMAs visible
    if (ch + 1 < NCHUNK) issue_chunk((ch + 1) * KC, (ch + 1) & 1);
    __builtin_prefetch(Arow + (size_t)(ch + 1) * KC, 0, 0);  // next A chunk -> cache

    const float* b0p = &bsm[ch & 1][lo  * KC + 2 * hi];
    const float* b1p = &bsm[ch & 1][n2c * KC + 2 * hi];
    const float* ap  = Arow + (size_t)ch * KC;
    for (int k0 = 0; k0 < KC; k0 += 4) {
      const v2f a  = *(const v2f*)(ap  + k0);   // A[m][ch*KC + k0 + 2*hi + j]
      const v2f b0 = *(const v2f*)(b0p + k0);   // cap[n][...] from LDS
      const v2f b1 = *(const v2f*)(b1p + k0);
      acc0 = __builtin_amdgcn_wmma_f32_16x16x4_f32(false, a, false, b0, (short)0, acc0, false, false);
      acc1 = __builtin_amdgcn_wmma_f32_16x16x4_f32(false, a, false, b1, (short)0, acc1, false, false);
    }
    __syncthreads();  // all waves done reading buf (ch&1) before it is re-filled
  }

  // leaky-ReLU + l2norm over the 24 valid w columns of each row
  for (int j = 0; j < 8; ++j) {
    float v0 = acc0[j]; v0 = v0 > 0.f ? v0 : 0.1f * v0;
    float v1 = acc1[j]; v1 = v1 > 0.f ? v1 : 0.1f * v1;
    const bool valid1 = lo < (N_WORD - 16);    // cols 16..23 only
    float ss = v0 * v0 + (valid1 ? v1 * v1 : 0.f);
    ss = half16_sum(ss);
    const float inv = 1.f / (sqrtf(ss) + EPSF);
    const int m = m0 + j + 8 * hi;             // global (b*36+r)
    const int b = m / N_REGION, r = m % N_REGION;
    float* dst = attn + ((size_t)(c * N_IMAGE + b) * N_WORD) * N_REGION;
    dst[(size_t)lo * N_REGION + r] = v0 * inv;
    if (valid1) dst[(size_t)(16 + lo) * N_REGION + r] = v1 * inv;
  }
}

// ======================================================================
// Kernel 4: softmax over regions (36) per (c,b,w) row, one wave per row
// ======================================================================
__global__ void softmax_r_kernel(float* __restrict__ attn) {
  const int wv   = threadIdx.x >> 5;
  const int lane = threadIdx.x & 31;
  const size_t row = (size_t)blockIdx.x * 8 + wv;
  float* p = attn + row * N_REGION;
  const bool hasHigh = lane < (N_REGION - 32);
  float x0 = LAMBDA_F * p[lane];
  float x1 = hasHigh ? LAMBDA_F * p[lane + 32] : -1e30f;
  float mx = fmaxf(x0, x1);
  for (int m = 16; m; m >>= 1) mx = fmaxf(mx, __shfl_xor(mx, m, 32));
  const float e0 = expf(x0 - mx);
  const float e1 = hasHigh ? expf(x1 - mx) : 0.f;
  float s = e0 + e1;
  for (int m = 16; m; m >>= 1) s += __shfl_xor(s, m, 32);
  p[lane] = e0 / s;
  if (hasHigh) p[lane + 32] = e1 / s;
}

// ======================================================================
// Kernel 5: wctx (fp32 WMMA, K=36) fused with block-cosine -> tnode.
//   wctx tiles live only in accumulators; 402 MB intermediate never stored.
// grid: 4096 (c*128+b) blocks, 256 threads = 8 waves * 8 tasks each
// ======================================================================
__global__ void wctx_tnode_kernel(const float* __restrict__ images,
                                  const float* __restrict__ captions,
                                  const float* __restrict__ attn,
                                  const float* __restrict__ qn,
                                  float* __restrict__ tnode) {
  const int cb = blockIdx.x;
  const int c = cb >> 7, b = cb & 127;
  const int wv   = threadIdx.x >> 5;
  const int lane = threadIdx.x & 31;
  const int lo = lane & 15, hi = lane >> 4;
  const float* attn_cb = attn + (size_t)cb * N_WORD * N_REGION;
  const float* img_b   = images + (size_t)b * N_REGION * EMBED;
  const float* cap     = captions + (size_t)c * N_WORD * EMBED;

  for (int t = 0; t < 8; ++t) {
    const int task = wv + 8 * t;       // 0..63
    const int mt = task >> 5;          // 0..1 : w tile
    const int kb = task & 31;          // 0..31: 32-wide d block
    const int m0 = mt * 16;
    const int d0 = kb * BLOCKW;
    const int wc = (m0 + lo < N_WORD) ? (m0 + lo) : (N_WORD - 1);  // clamp pad rows
    const float* Arow = attn_cb + (size_t)wc * N_REGION + 2 * hi;

    v8f acc0 = {}; v8f acc1 = {};
    for (int k0 = 0; k0 < N_REGION; k0 += 4) {       // K = 36 = 9 * 4
      const v2f a = *(const v2f*)(Arow + k0);        // attnS[w][r]
      const int r = k0 + 2 * hi;
      v2f b0, b1;
      b0[0] = img_b[(size_t)r       * EMBED + d0 + lo];
      b0[1] = img_b[(size_t)(r + 1) * EMBED + d0 + lo];
      b1[0] = img_b[(size_t)r       * EMBED + d0 + 16 + lo];
      b1[1] = img_b[(size_t)(r + 1) * EMBED + d0 + 16 + lo];
      acc0 = __builtin_amdgcn_wmma_f32_16x16x4_f32(false, a, false, b0, (short)0, acc0, false, false);
      acc1 = __builtin_amdgcn_wmma_f32_16x16x4_f32(false, a, false, b1, (short)0, acc1, false, false);
    }
    // block cosine against q = cap, straight from accumulators
    for (int j = 0; j < 8; ++j) {
      const int w = m0 + j + 8 * hi;
      const bool valid = w < N_WORD;
      const int wq = valid ? w : 0;
      const float qv0 = cap[(size_t)wq * EMBED + d0 + lo];
      const float qv1 = cap[(size_t)wq * EMBED + d0 + 16 + lo];
      const float v0 = acc0[j], v1 = acc1[j];
      float num = qv0 * v0 + qv1 * v1;
      float ssq = v0 * v0 + v1 * v1;
      num = half16_sum(num);
      ssq = half16_sum(ssq);
      if (valid && lo == 0) {
        const float cn  = sqrtf(ssq);
        const float qnv = qn[(size_t)(c * N_WORD + w) * NUM_BLOCK + kb];
        const float den = fmaxf(qnv * cn, EPSF);
        tnode[((size_t)cb * N_WORD + w) * NUM_BLOCK + kb] = num / den;
      }
    }
  }
}

// ======================================================================
// Kernel 6: graph conv + MLP + mean over words; one block per (c,b)
// ======================================================================
__global__ void graph_mlp_kernel(const float* __restrict__ tnode,
                                 const float* __restrict__ kwa,
                                 const float* __restrict__ conv_w,
                                 const float* __restrict__ W1n,
                                 const float* __restrict__ b1,
                                 const float* __restrict__ W2n,
                                 const float* __restrict__ b2,
                                 float* __restrict__ out) {
  __shared__ float kwa_s[N_WORD * N_WORD * N_KERNELS];               // 4608
  __shared__ float tn_s[N_WORD * NUM_BLOCK];                         // 768
  __shared__ float conv_s[N_KERNELS * NUM_BLOCK * (HID / N_KERNELS)];// 2048
  __shared__ float w1_s[HID * HID];                                  // 4096
  __shared__ float agg_s[256];
  __shared__ float hid_s[HID];
  __shared__ float h_s[HID];
  __shared__ float red_s[HID];
  __shared__ float s_sum;
  const int cb = blockIdx.x;
  const int c = cb >> 7, b = cb & 127;
  const int tid = threadIdx.x;

  for (int p = tid; p < N_WORD * N_WORD * N_KERNELS; p += blockDim.x)
    kwa_s[p] = kwa[(size_t)c * N_WORD * N_WORD * N_KERNELS + p];
  for (int p = tid; p < N_WORD * NUM_BLOCK; p += blockDim.x)
    tn_s[p] = tnode[(size_t)cb * N_WORD * NUM_BLOCK + p];
  for (int p = tid; p < N_KERNELS * NUM_BLOCK * (HID / N_KERNELS); p += blockDim.x)
    conv_s[p] = conv_w[p];
  for (int p = tid; p < HID * HID; p += blockDim.x) w1_s[p] = W1n[p];
  if (tid == 0) s_sum = 0.f;
  __syncthreads();

  const int k = tid >> 5, f = tid & 31;  // (kernel, feature) for agg
  for (int i = 0; i < N_WORD; ++i) {
    // agg[k][f] = sum_j kwa[i][j][k] * tnode[j][f]
    float a = 0.f;
    const float* kr = kwa_s + (size_t)(i * N_WORD) * N_KERNELS + k;
    for (int j = 0; j < N_WORD; ++j) a += kr[(size_t)j * N_KERNELS] * tn_s[j * NUM_BLOCK + f];
    agg_s[tid] = a;
    __syncthreads();
    // hidden[k][o] = sum_f agg[k][f] * conv_w[k][f][o]
    if (tid < HID) {
      const int kk = tid >> 3, o = tid & 7;
      float hsum = 0.f;
      for (int ff = 0; ff < NUM_BLOCK; ++ff)
        hsum += agg_s[kk * NUM_BLOCK + ff] * conv_s[(kk * NUM_BLOCK + ff) * (HID / N_KERNELS) + o];
      hid_s[tid] = hsum;
    }
    __syncthreads();
    // h = tanh(hidden @ W1n^T + b1)
    if (tid < HID) {
      float hv = b1[tid];
      for (int u = 0; u < HID; ++u) hv += w1_s[tid * HID + u] * hid_s[u];
      h_s[tid] = tanhf(hv);
    }
    __syncthreads();
    if (tid < HID) red_s[tid] = W2n[tid] * h_s[tid];
    __syncthreads();
    if (tid == 0) {
      float s = b2[0];
      for (int u = 0; u < HID; ++u) s += red_s[u];
      s_sum += s;
    }
    __syncthreads();
  }
  if (tid == 0) out[(size_t)b * N_CAPTION + c] = s_sum / (float)N_WORD;
}

// ======================================================================
extern "C" void kernel_launch(void* const* d_in, const int* in_sizes, int n_in,
                              void* d_out, int out_size, void* d_ws, size_t ws_size,
                              hipStream_t stream) {
  const float* images   = (const float*)d_in[0];
  const float* captions = (const float*)d_in[1];
  const int*   depends  = (const int*)d_in[2];
  // d_in[3] = cap_lens (all == N_WORD, unused)
  const float* mean_k = (const float*)d_in[4];
  const float* prec_k = (const float*)d_in[5];
  const float* conv_w = (const float*)d_in[6];
  const float* out1_v = (const float*)d_in[7];
  const float* out1_g = (const float*)d_in[8];
  const float* out1_b = (const float*)d_in[9];
  const float* out2_v = (const float*)d_in[10];
  const float* out2_g = (const float*)d_in[11];
  const float* out2_b = (const float*)d_in[12];
  float* out = (float*)d_out;

  float* ws    = (float*)d_ws;   // ~27.5 MB of fp32 scratch used
  float* attn  = ws + OFF_ATTN;
  float* tnode = ws + OFF_TNODE;
  float* kwa   = ws + OFF_KWA;
  float* qn    = ws + OFF_QN;
  float* W1n   = ws + OFF_W1N;
  float* W2n   = ws + OFF_W2N;

  weight_prep_kernel<<<1, 64, 0, stream>>>(out1_v, out1_g, out2_v, out2_g, W1n, W2n);
  caption_prep_kernel<<<N_CAPTION, 256, 0, stream>>>(captions, depends, mean_k, prec_k, kwa, qn);
  attn_gemm_kernel<<<N_CAPTION * (288 / 8), 256, 0, stream>>>(images, captions, attn);
  softmax_r_kernel<<<(N_CAPTION * N_IMAGE * N_WORD) / 8, 256, 0, stream>>>(attn);
  wctx_tnode_kernel<<<N_CAPTION * N_IMAGE, 256, 0, stream>>>(images, captions, attn, qn, tnode);
  graph_mlp_kernel<<<N_CAPTION * N_IMAGE, 256, 0, stream>>>(tnode, kwa, conv_w, W1n, out1_b, W2n, out2_b, out);
}